// SGC_new_40544491274370
// MI455X (gfx1250) — compile-verified
//
#include <hip/hip_runtime.h>
#include <hip/hip_bf16.h>

#define N_NODES   100000
#define N_EDGES   1600000
#define N_FEATS   128
#define N_CLASSES 40
#define N_PAD     48      // 3 tiles of 16 covering 40 output cols

typedef float v2f __attribute__((ext_vector_type(2)));
typedef float v8f __attribute__((ext_vector_type(8)));

// ---------------- utility: zero a float buffer ----------------
__global__ void __launch_bounds__(256) sgc_zero_f32(float* __restrict__ p, int n) {
    int i = blockIdx.x * 256 + threadIdx.x;
    if (i < n) p[i] = 0.0f;
}

// ---------------- deg[col[e]] += w[e] ----------------
__global__ void __launch_bounds__(256) sgc_deg(const long long* __restrict__ col,
                                               const float* __restrict__ w,
                                               float* __restrict__ deg) {
    int e = blockIdx.x * 256 + threadIdx.x;
    if (e < N_EDGES) atomicAdd(&deg[col[e]], w[e]);
}

// ---------------- deg_inv_sqrt ----------------
__global__ void __launch_bounds__(256) sgc_dis(const float* __restrict__ deg,
                                               float* __restrict__ dis) {
    int i = blockIdx.x * 256 + threadIdx.x;
    if (i < N_NODES) {
        float d = deg[i];
        dis[i] = (d > 0.0f) ? rsqrtf(fmaxf(d, 1e-30f)) : 0.0f;
    }
}

// ---------------- norm[e] = dis[row]*w*dis[col] ----------------
__global__ void __launch_bounds__(256) sgc_norm(const long long* __restrict__ row,
                                                const long long* __restrict__ col,
                                                const float* __restrict__ w,
                                                const float* __restrict__ dis,
                                                float* __restrict__ norm) {
    int e = blockIdx.x * 256 + threadIdx.x;
    if (e < N_EDGES) norm[e] = dis[row[e]] * w[e] * dis[col[e]];
}

// ---------------- zero-pad + transpose W: WpadT[n][k] = W[k][n] (n<40), else 0 ----
__global__ void __launch_bounds__(256) sgc_padW(const float* __restrict__ Wm,
                                                float* __restrict__ WpadT) {
    int i = blockIdx.x * 256 + threadIdx.x;           // i = n*128 + k
    if (i < N_PAD * N_FEATS) {
        int n = i / N_FEATS;
        int k = i % N_FEATS;
        WpadT[i] = (n < N_CLASSES) ? Wm[(size_t)k * N_CLASSES + n] : 0.0f;
    }
}

// ---------------- one propagation hop ----------------
// One wave (32 lanes) per edge; each lane moves a float4 (32*4 = 128 feats).
// h (51 MB) is L2-resident on MI455X (192 MB L2), so gathers + atomics hit L2.
__global__ void __launch_bounds__(256) sgc_hop(const float* __restrict__ h_in,
                                               const long long* __restrict__ row,
                                               const long long* __restrict__ col,
                                               const float* __restrict__ norm,
                                               float* __restrict__ h_out) {
    int t = blockIdx.x * 256 + threadIdx.x;
    int e = t >> 5;
    if (e >= N_EDGES) return;
    int lane = t & 31;
    long long r = row[e];
    long long c = col[e];
    float s = norm[e];
    const float4 v = *(const float4*)(h_in + (size_t)r * N_FEATS + lane * 4);
    float* dst = h_out + (size_t)c * N_FEATS + lane * 4;
    atomicAdd(dst + 0, v.x * s);
    atomicAdd(dst + 1, v.y * s);
    atomicAdd(dst + 2, v.z * s);
    atomicAdd(dst + 3, v.w * s);
}

// ---------------- final linear: out = h @ W via V_WMMA_F32_16X16X4_F32 ----------------
// One wave computes one 16x16 D tile. M tiles = 100000/16 = 6250 (exact).
// N tiles = 3 (48 cols, zero-padded W so no conditionals in the k-loop).
// EXEC stays all-1s through the WMMA loop; inner loop = 2x global_load_b64 + wmma.
__global__ void __launch_bounds__(64) sgc_gemm_wmma(const float* __restrict__ H,
                                                    const float* __restrict__ WpadT,
                                                    float* __restrict__ out) {
    const int NT_N = 3;
    int wave  = (blockIdx.x * 64 + threadIdx.x) >> 5;   // exactly 18750 waves
    int mTile = wave / NT_N;
    int nTile = wave % NT_N;
    int lane  = threadIdx.x & 31;
    int half  = lane >> 4;       // 0 = lanes 0-15, 1 = lanes 16-31
    int l16   = lane & 15;

    // A (16x4 f32): lane half 0 holds K%4 in {0,1}; half 1 holds {2,3}
    const float* Arow = H + ((size_t)mTile * 16 + l16) * N_FEATS + half * 2;
    // B (4x16 f32): same K-split per lane half; column-major padded W
    int bcol = nTile * 16 + l16;
    const float* Brow = WpadT + (size_t)bcol * N_FEATS + half * 2;

    v8f c = {0.f, 0.f, 0.f, 0.f, 0.f, 0.f, 0.f, 0.f};
    for (int kk = 0; kk < N_FEATS; kk += 4) {
        v2f a, b;
        a.x = Arow[kk];
        a.y = Arow[kk + 1];
        b.x = Brow[kk];
        b.y = Brow[kk + 1];
        // 8 args: (neg_a, A, neg_b, B, c_mod, C, reuse_a, reuse_b)
        c = __builtin_amdgcn_wmma_f32_16x16x4_f32(false, a, false, b,
                                                  (short)0, c, false, false);
    }

    // D layout: VGPR r -> row (half*8 + r), col = l16
    size_t rowBase = (size_t)mTile * 16 + half * 8;
    if (bcol < N_CLASSES) {
        #pragma unroll
        for (int r = 0; r < 8; ++r)
            out[(rowBase + r) * N_CLASSES + bcol] = c[r];
    }
}

extern "C" void kernel_launch(void* const* d_in, const int* in_sizes, int n_in,
                              void* d_out, int out_size, void* d_ws, size_t ws_size,
                              hipStream_t stream) {
    const float*     x   = (const float*)d_in[0];
    const long long* ei  = (const long long*)d_in[1];   // int64 [2, E]
    const float*     ew  = (const float*)d_in[2];
    const float*     Wm  = (const float*)d_in[3];
    float*           out = (float*)d_out;

    const long long* row = ei;
    const long long* col = ei + N_EDGES;

    // workspace layout (floats): deg[N] | dis[N] | norm[E] | hA[N*F] | hB[N*F] | WpadT[48*128]
    float* ws    = (float*)d_ws;
    float* deg   = ws;
    float* dis   = deg + N_NODES;
    float* norm  = dis + N_NODES;
    float* hA    = norm + N_EDGES;
    float* hB    = hA + (size_t)N_NODES * N_FEATS;
    float* WpadT = hB + (size_t)N_NODES * N_FEATS;

    const int ZB_N  = (N_NODES + 255) / 256;
    const int ZB_E  = (N_EDGES + 255) / 256;
    const int ZB_NF = (N_NODES * N_FEATS + 255) / 256;
    const int HOPB  = (N_EDGES * 32) / 256;             // 200000 blocks

    // gcn_norm
    sgc_zero_f32<<<ZB_N, 256, 0, stream>>>(deg, N_NODES);
    sgc_deg     <<<ZB_E, 256, 0, stream>>>(col, ew, deg);
    sgc_dis     <<<ZB_N, 256, 0, stream>>>(deg, dis);
    sgc_norm    <<<ZB_E, 256, 0, stream>>>(row, col, ew, dis, norm);

    // pad/transpose W for branch-free WMMA B loads
    sgc_padW<<<(N_PAD * N_FEATS + 255) / 256, 256, 0, stream>>>(Wm, WpadT);

    // 3 hops, ping-pong hA/hB  (x -> hA -> hB -> hA)
    sgc_zero_f32<<<ZB_NF, 256, 0, stream>>>(hA, N_NODES * N_FEATS);
    sgc_hop     <<<HOPB, 256, 0, stream>>>(x,  row, col, norm, hA);
    sgc_zero_f32<<<ZB_NF, 256, 0, stream>>>(hB, N_NODES * N_FEATS);
    sgc_hop     <<<HOPB, 256, 0, stream>>>(hA, row, col, norm, hB);
    sgc_zero_f32<<<ZB_NF, 256, 0, stream>>>(hA, N_NODES * N_FEATS);
    sgc_hop     <<<HOPB, 256, 0, stream>>>(hB, row, col, norm, hA);

    // final linear via fp32 WMMA: 6250 M-tiles * 3 N-tiles = 18750 waves = 9375 blocks * 2
    sgc_gemm_wmma<<<9375, 64, 0, stream>>>(hA, WpadT, out);
}